// global_gated_update_33320356282900
// MI455X (gfx1250) — compile-verified
//
#include <hip/hip_runtime.h>

// ---------------------------------------------------------------------------
// global_gated_update for MI455X (gfx1250)
//
// out[b, item, f] = emb_weight[item, f]                                (bulk)
// out[b, nodes[i], f] = (1-alpha)*emb_weight + alpha*nodes_output      (scatter)
//
// Pure bandwidth problem: 204.8 MB stores, ~26 MB loads -> ~10us at 23.3 TB/s.
// No matmul -> WMMA inapplicable. Use CDNA5 async LDS<->global data movers
// (ASYNCcnt path) for the 8-way broadcast: read emb_weight from HBM once into
// LDS, fan out 8 coalesced b128 stores per tile.
// ---------------------------------------------------------------------------

#define ITEMS_TOTAL 100000
#define FEAT 64
#define NBATCH 8

typedef float v4f __attribute__((ext_vector_type(4)));
typedef int   v4i __attribute__((ext_vector_type(4)));
// Builtin parameter types (from hipcc diagnostic): int4 in AS1 (global /
// __device__) and int4 in AS3 (LDS).
typedef __attribute__((address_space(1))) v4i* g4p;
typedef __attribute__((address_space(3))) v4i* l4p;

#if defined(__gfx1250__) && \
    __has_builtin(__builtin_amdgcn_global_load_async_to_lds_b128) && \
    __has_builtin(__builtin_amdgcn_global_store_async_from_lds_b128)
#define USE_ASYNC_LDS 1
#else
#define USE_ASYNC_LDS 0
#endif

__device__ __forceinline__ void wait_async_zero() {
#if __has_builtin(__builtin_amdgcn_s_wait_asynccnt)
  __builtin_amdgcn_s_wait_asynccnt(0);
#else
  asm volatile("s_wait_asynccnt 0" ::: "memory");
#endif
}

// ---------------------------------------------------------------------------
// Kernel 1: broadcast emb_weight into all 8 batch copies of out.
// One thread = one 16-byte chunk. n4 = ITEMS_TOTAL*FEAT/4 chunks.
// ---------------------------------------------------------------------------
__global__ __launch_bounds__(256) void bcast_emb_kernel(const v4f* __restrict__ emb4,
                                                        v4f* __restrict__ out4,
                                                        long long n4) {
  const int tid = threadIdx.x;
  const long long idx = (long long)blockIdx.x * 256 + tid;
  if (idx >= n4) return;

#if USE_ASYNC_LDS
  __shared__ v4f tile[256];
  // Stage 16B/lane from global into LDS via the async data mover (ASYNCcnt).
  __builtin_amdgcn_global_load_async_to_lds_b128(
      (g4p)(emb4 + idx), (l4p)(&tile[tid]), /*offset=*/0, /*cpol=*/0);
  wait_async_zero();  // load landed in LDS; each lane only touches its own 16B
#pragma unroll
  for (int b = 0; b < NBATCH; ++b) {
    __builtin_amdgcn_global_store_async_from_lds_b128(
        (g4p)(out4 + (long long)b * n4 + idx), (l4p)(&tile[tid]),
        /*offset=*/0, /*cpol=*/0);
  }
  wait_async_zero();  // drain before wave exit / LDS reuse
#else
  v4f v = emb4[idx];
#pragma unroll
  for (int b = 0; b < NBATCH; ++b) {
    // Streaming 205MB write: non-temporal so we don't evict useful L2 lines.
    __builtin_nontemporal_store(v, out4 + (long long)b * n4 + idx);
  }
#endif
}

// ---------------------------------------------------------------------------
// Kernel 2: scatter the gated updates over the touched rows.
// One thread = one output float. total = n_nodes * FEAT threads.
// ---------------------------------------------------------------------------
__global__ __launch_bounds__(256) void scatter_update_kernel(
    const float* __restrict__ nodes_output,  // [n_nodes, FEAT]
    const float* __restrict__ emb_weight,    // [ITEMS_TOTAL, FEAT]
    const float* __restrict__ alpha,         // [ITEMS_TOTAL]
    const int* __restrict__ nodes,           // [n_nodes]
    const int* __restrict__ ptr,             // [n_batch+1]
    float* __restrict__ out,                 // [n_batch, ITEMS_TOTAL, FEAT]
    int n_nodes, int n_batch) {
  const long long t = (long long)blockIdx.x * blockDim.x + threadIdx.x;
  const long long total = (long long)n_nodes * FEAT;
  if (t >= total) return;

  const int i = (int)(t >> 6);   // node index (FEAT == 64)
  const int f = (int)(t & 63);   // feature index

  // batch id: ptr[b] <= i < ptr[b+1]; B is tiny (8), linear scan is fine and
  // ptr lives in L2/scalar cache.
  int b = 0;
  while (b + 1 < n_batch && ptr[b + 1] <= i) ++b;

  const int item = nodes[i];
  const float a = alpha[item];
  const float v = (1.0f - a) * emb_weight[(long long)item * FEAT + f] +
                  a * nodes_output[(long long)i * FEAT + f];
  out[((long long)b * ITEMS_TOTAL + item) * FEAT + f] = v;
}

// ---------------------------------------------------------------------------
// Host launcher
// inputs: 0 nodes_output f32[4096*64], 1 emb_weight f32[100000*64],
//         2 alpha f32[100000], 3 nodes i32[4096], 4 ptr i32[9]
// output: f32[8*100000*64]
// ---------------------------------------------------------------------------
extern "C" void kernel_launch(void* const* d_in, const int* in_sizes, int n_in,
                              void* d_out, int out_size, void* d_ws, size_t ws_size,
                              hipStream_t stream) {
  (void)n_in; (void)out_size; (void)d_ws; (void)ws_size;

  const float* nodes_output = (const float*)d_in[0];
  const float* emb_weight   = (const float*)d_in[1];
  const float* alpha        = (const float*)d_in[2];
  const int*   nodes        = (const int*)d_in[3];
  const int*   ptr          = (const int*)d_in[4];
  float*       out          = (float*)d_out;

  const int n_nodes = in_sizes[3];
  const int n_batch = in_sizes[4] - 1;

  // ---- Kernel 1: 8-way broadcast of emb_weight (bandwidth-bound) ----
  const long long n4 = (long long)ITEMS_TOTAL * FEAT / 4;  // 1,600,000 chunks
  const unsigned g1 = (unsigned)((n4 + 255) / 256);        // 6250 blocks
  bcast_emb_kernel<<<dim3(g1), dim3(256), 0, stream>>>(
      (const v4f*)emb_weight, (v4f*)out, n4);

  // ---- Kernel 2: scatter gated rows (tiny, 1 MB of traffic) ----
  const long long tot = (long long)n_nodes * FEAT;
  const unsigned g2 = (unsigned)((tot + 255) / 256);
  scatter_update_kernel<<<dim3(g2), dim3(256), 0, stream>>>(
      nodes_output, emb_weight, alpha, nodes, ptr, out, n_nodes, n_batch);
}